// PropNetDensityFieldMS_17119739642225
// MI455X (gfx1250) — compile-verified
//
#include <hip/hip_runtime.h>
#include <math.h>

typedef float v2f __attribute__((ext_vector_type(2)));
typedef float v8f __attribute__((ext_vector_type(8)));

#define TBITS 18
#define TSIZE (1u << TBITS)
#define KF    8
#define LVL   8
#define HID   64
#define BLK   256
#define ESTR  17          // 16 features + 1 pad to break LDS bank conflicts

__device__ __forceinline__ v8f wmma4(v2f a, v2f b, v8f c) {
  // V_WMMA_F32_16X16X4_F32: D(16x16,f32) = A(16x4,f32) * B(4x16,f32) + C
  return __builtin_amdgcn_wmma_f32_16x16x4_f32(false, a, false, b, (short)0, c,
                                               false, false);
}

__device__ __forceinline__ void l2acc(v8f c, float bias, float w2, float* acc) {
#pragma unroll
  for (int r = 0; r < 8; ++r) {
    float h = c[r] + bias;
    h = fmaxf(h, 0.0f);
    acc[r] += h * w2;
  }
}

__global__ __launch_bounds__(BLK) void propnet_density_kernel(
    const float* __restrict__ positions,  // [N,3]
    const float* __restrict__ centroids,  // [K,3]
    const float* __restrict__ tables,     // [K,L,T,F] f32
    const float* __restrict__ W1,         // [K,16,64]
    const float* __restrict__ b1,         // [K,64]
    const float* __restrict__ W2,         // [K,64,1]
    const float* __restrict__ b2,         // [K,1]
    float* __restrict__ out,              // [N,1]
    int N) {
  __shared__ float          encS[BLK * ESTR];
  __shared__ unsigned short list[KF * BLK];
  __shared__ unsigned int   cnt[KF];

  const int tid   = threadIdx.x;
  const int gbase = blockIdx.x * BLK;
  const int g     = gbase + tid;

  if (tid < KF) cnt[tid] = 0u;
#pragma unroll
  for (int i = 0; i < KF; ++i) list[i * BLK + tid] = 0xFFFFu;
  __syncthreads();

  // ---------------- Phase 1: cluster assign + hashgrid encode ----------------
  int myk = -1;
  if (g < N) {
    const float px = positions[3 * g + 0];
    const float py = positions[3 * g + 1];
    const float pz = positions[3 * g + 2];

    float best = 3.4e38f;
#pragma unroll
    for (int kk = 0; kk < KF; ++kk) {
      float dx = px - centroids[3 * kk + 0];
      float dy = py - centroids[3 * kk + 1];
      float dz = pz - centroids[3 * kk + 2];
      float d2 = dx * dx + dy * dy + dz * dz;
      if (d2 < best) { best = d2; myk = kk; }   // strict < == first-index tie-break
    }

    const float2* tab = (const float2*)tables;  // F=2 packed
    const size_t  kbase = (size_t)(unsigned)myk * (size_t)LVL * (size_t)TSIZE;
    for (int l = 0; l < LVL; ++l) {
      const float res = (float)(16 << l);
      const float sx = px * res, sy = py * res, sz = pz * res;
      const float fx = floorf(sx), fy = floorf(sy), fz = floorf(sz);
      const float wx1 = sx - fx, wy1 = sy - fy, wz1 = sz - fz;
      const float wx0 = 1.0f - wx1, wy0 = 1.0f - wy1, wz0 = 1.0f - wz1;
      const unsigned ix = (unsigned)fx, iy = (unsigned)fy, iz = (unsigned)fz;
      const size_t lbase = kbase + (size_t)l * (size_t)TSIZE;

      float a0 = 0.0f, a1 = 0.0f;
#pragma unroll
      for (int c = 0; c < 8; ++c) {
        const unsigned cx = ix + (unsigned)((c >> 2) & 1);
        const unsigned cy = iy + (unsigned)((c >> 1) & 1);
        const unsigned cz = iz + (unsigned)(c & 1);
        unsigned h = cx ^ (cy * 2654435761u) ^ (cz * 805459861u);
        h &= (TSIZE - 1u);
        const float2 f2 = tab[lbase + h];               // L2-resident gather
        const float  cw = ((c & 4) ? wx1 : wx0) *
                          ((c & 2) ? wy1 : wy0) *
                          ((c & 1) ? wz1 : wz0);
        a0 += f2.x * cw;
        a1 += f2.y * cw;
      }
      encS[tid * ESTR + 2 * l + 0] = a0;
      encS[tid * ESTR + 2 * l + 1] = a1;
    }
  }

  // ---------------- Phase 2: in-block counting-sort by cluster ---------------
  if (myk >= 0) {
    unsigned r = atomicAdd(&cnt[myk], 1u);
    list[myk * BLK + r] = (unsigned short)tid;
  }
  __syncthreads();

  // Tile schedule (uniform across the block): tiles of 16 points, per cluster.
  unsigned pre[KF + 1];
  pre[0] = 0;
#pragma unroll
  for (int kk = 0; kk < KF; ++kk) pre[kk + 1] = pre[kk] + ((cnt[kk] + 15u) >> 4);
  const unsigned totalTiles = pre[KF];

  const int wave = tid >> 5;
  const int lane = tid & 31;
  const int m    = lane & 15;   // column / point-row selector
  const int half = lane >> 4;   // 0: lanes 0-15, 1: lanes 16-31

  // ---------------- Phase 3: WMMA MLP per cluster-uniform tile ---------------
  for (unsigned tile = wave; tile < totalTiles; tile += (BLK / 32)) {
    int k = 0;
    unsigned j = 0;
#pragma unroll
    for (int kk = 0; kk < KF; ++kk)
      if (tile >= pre[kk] && tile < pre[kk + 1]) { k = kk; j = tile - pre[kk]; }

    const float* W1k = W1 + k * (16 * HID);
    const float* b1k = b1 + k * HID;
    const float* W2k = W2 + k * HID;
    const float  b2v = b2[k];

    const unsigned short lt   = list[k * BLK + j * 16 + m];
    const unsigned       ltid = (lt == 0xFFFFu) ? 0u : (unsigned)lt; // pad rows read row 0 (independent rows)
    const float*         er   = &encS[ltid * ESTR];

    v8f c0 = {}, c1 = {}, c2 = {}, c3 = {};
#pragma unroll
    for (int s = 0; s < 4; ++s) {
      v2f a;
      a.x = er[4 * s + 2 * half + 0];   // A layout: VGPR0 -> K = 4s + 2*half
      a.y = er[4 * s + 2 * half + 1];   //           VGPR1 -> K = 4s + 2*half + 1
      const float* w0 = W1k + (4 * s + 2 * half) * HID;
      const float* w1 = w0 + HID;
      v2f b;
      b.x = w0[0 + m];  b.y = w1[0 + m];
      c0 = wmma4(a, b, c0);
      b.x = w0[16 + m]; b.y = w1[16 + m];
      c1 = wmma4(a, b, c1);
      b.x = w0[32 + m]; b.y = w1[32 + m];
      c2 = wmma4(a, b, c2);
      b.x = w0[48 + m]; b.y = w1[48 + m];
      c3 = wmma4(a, b, c3);
    }

    // Layer 2: relu(h + b1) dot W2, reduced across the 16 columns of each half.
    float acc[8];
#pragma unroll
    for (int r = 0; r < 8; ++r) acc[r] = 0.0f;
    l2acc(c0, b1k[0 + m],  W2k[0 + m],  acc);
    l2acc(c1, b1k[16 + m], W2k[16 + m], acc);
    l2acc(c2, b1k[32 + m], W2k[32 + m], acc);
    l2acc(c3, b1k[48 + m], W2k[48 + m], acc);

#pragma unroll
    for (int r = 0; r < 8; ++r) {
      acc[r] += __shfl_xor(acc[r], 1, 16);
      acc[r] += __shfl_xor(acc[r], 2, 16);
      acc[r] += __shfl_xor(acc[r], 4, 16);
      acc[r] += __shfl_xor(acc[r], 8, 16);
    }

    if (m == 0) {  // lane 0 writes rows 0..7, lane 16 writes rows 8..15
#pragma unroll
      for (int r = 0; r < 8; ++r) {
        const int            row = r + 8 * half;
        const unsigned short l2  = list[k * BLK + j * 16 + row];
        if (l2 != 0xFFFFu) out[gbase + l2] = expf(acc[r] + b2v);
      }
    }
  }
}

extern "C" void kernel_launch(void* const* d_in, const int* in_sizes, int n_in,
                              void* d_out, int out_size, void* d_ws, size_t ws_size,
                              hipStream_t stream) {
  const float* positions = (const float*)d_in[0];
  const float* centroids = (const float*)d_in[1];
  const float* tables    = (const float*)d_in[2];
  const float* W1        = (const float*)d_in[3];
  const float* b1        = (const float*)d_in[4];
  const float* W2        = (const float*)d_in[5];
  const float* b2        = (const float*)d_in[6];
  float*       out       = (float*)d_out;

  const int N = in_sizes[0] / 3;
  const int grid = (N + BLK - 1) / BLK;
  propnet_density_kernel<<<grid, BLK, 0, stream>>>(
      positions, centroids, tables, W1, b1, W2, b2, out, N);
}